// MPA_37056977830474
// MI455X (gfx1250) — compile-verified
//
#include <hip/hip_runtime.h>

typedef float v2f __attribute__((ext_vector_type(2)));
typedef float v8f __attribute__((ext_vector_type(8)));

// One wave32 does the whole problem (M=4, F=4, V=6).
// The three einsum contractions are computed as three chains of four
// V_WMMA_F32_16X16X4_F32 (exact fp32), i.e. D(16x16) = W(16x16) * X(16x16)
// with K split into 4 steps of 4.
__global__ __launch_bounds__(32)
void MPA_37056977830474_kernel(const float* __restrict__ IVF,
                               const int*   __restrict__ VN_index,
                               const int*   __restrict__ m_s,
                               const int*   __restrict__ n_s,
                               const int*   __restrict__ FN_index,
                               const float* __restrict__ fa_n,
                               const float* __restrict__ w0,
                               float*       __restrict__ out)
{
    __shared__ float sIVF[96];        // IVF[m][f][v]  ((m*4+f)*6+v)
    __shared__ float sW0[96];
    __shared__ float sFA[256];        // fa[f][i][j][k]
    __shared__ int   sFN[12];         // FN_index[f][j]
    __shared__ int   sVN[12];         // [va[0..5], vb[0..5]]
    __shared__ float sG[3][4][4];     // g_j[m][f] = IVF[m,f,FN[f][j]]
    __shared__ float sA[3][64];       // sA[j][(4f+a)*4 + col]  (D cols 0..3)

    const int lane = threadIdx.x;     // 0..31, single wave

    // ---- stage inputs to LDS -------------------------------------------
    for (int i = lane; i < 96; i += 32) { sIVF[i] = IVF[i]; sW0[i] = w0[i]; }
    for (int i = lane; i < 256; i += 32) sFA[i] = fa_n[i];
    if (lane < 12) { sFN[lane] = FN_index[lane]; sVN[lane] = VN_index[lane]; }
    __syncthreads();

    // ---- gather g0,g1,g2 -----------------------------------------------
    if (lane < 16) {
        const int q = lane >> 2, f = lane & 3;
        for (int j = 0; j < 3; ++j)
            sG[j][q][f] = sIVF[(q * 4 + f) * 6 + sFN[f * 3 + j]];
    }
    __syncthreads();

    // ---- WMMA chains ----------------------------------------------------
    // A operand (f32 16x4): lane holds M=lane%16; reg0 -> K=2*half, reg1 -> K=2*half+1
    // B operand (f32 4x16): lanes 0-15 hold K=0,1; lanes 16-31 hold K=2,3; N=lane%16
    // D (f32 16x16): reg i, lane L -> M = i + 8*half, N = L%16
    const int row  = lane & 15;
    const int half = lane >> 4;
    const int f_r  = row >> 2;        // A-row factor index
    const int a_r  = row & 3;         // A-row output index
    const int col  = lane & 15;       // D/B column
    const int fc   = col & 3;         // column's f' (cols>=4 unused duplicates)
    const int kk0  = 2 * half;
    const int kk1  = 2 * half + 1;

    #pragma unroll
    for (int j = 0; j < 3; ++j) {
        v8f acc = {0.f, 0.f, 0.f, 0.f, 0.f, 0.f, 0.f, 0.f};
        #pragma unroll
        for (int s = 0; s < 4; ++s) {
            // W_j[row, k=4s+kk] : transpositions of fa
            float a0, a1, b0, b1;
            if (j == 0) {              // A0: fa[f, c=s, a, b=kk], X = g2[s]*g1[kk]
                a0 = sFA[((f_r * 4 + s) * 4 + a_r) * 4 + kk0];
                a1 = sFA[((f_r * 4 + s) * 4 + a_r) * 4 + kk1];
                b0 = sG[2][s][fc] * sG[1][kk0][fc];
                b1 = sG[2][s][fc] * sG[1][kk1][fc];
            } else if (j == 1) {       // A1: fa[f, c=s, b=kk, a], X = g2[s]*g0[kk]
                a0 = sFA[((f_r * 4 + s) * 4 + kk0) * 4 + a_r];
                a1 = sFA[((f_r * 4 + s) * 4 + kk1) * 4 + a_r];
                b0 = sG[2][s][fc] * sG[0][kk0][fc];
                b1 = sG[2][s][fc] * sG[0][kk1][fc];
            } else {                   // A2: fa[f, a, b=s, c=kk], X = g0[s]*g1[kk]
                a0 = sFA[((f_r * 4 + a_r) * 4 + s) * 4 + kk0];
                a1 = sFA[((f_r * 4 + a_r) * 4 + s) * 4 + kk1];
                b0 = sG[0][s][fc] * sG[1][kk0][fc];
                b1 = sG[0][s][fc] * sG[1][kk1][fc];
            }
            v2f av = {a0, a1};
            v2f bv = {b0, b1};
            acc = __builtin_amdgcn_wmma_f32_16x16x4_f32(
                      /*neg_a=*/false, av, /*neg_b=*/false, bv,
                      /*c_mod=*/(short)0, acc,
                      /*reuse_a=*/false, /*reuse_b=*/false);
        }
        // scatter D columns 0..3 (the only meaningful ones) to LDS
        if (col < 4) {
            #pragma unroll
            for (int i = 0; i < 8; ++i)
                sA[j][(i + half * 8) * 4 + col] = acc[i];
        }
        __syncthreads();
    }

    // ---- epilogue: IFV scatter-add + mean + ratio update + *w0 ----------
    const int mm = m_s[0];
    const int nn = n_s[0];
    for (int t = lane; t < 96; t += 32) {
        const int m   = t / 24;
        const int rem = t % 24;
        const int f   = rem / 6;
        const int v   = rem % 6;
        float res;
        if (mm >= nn) {
            res = sIVF[t];                       // reference: return IVF as-is
        } else {
            const int va = sVN[v];
            const int vb = sVN[6 + v];
            if (f == va || f == vb) {
                const int fo = (f == va) ? vb : va;     // the "other" factor
                // IFV[m,fo,v] == A_j[m,fo] where FN[fo][j] == v (unique j)
                int jj = 0;
                for (int j = 0; j < 3; ++j)
                    if (sFN[fo * 3 + j] == v) jj = j;
                const float ifv = sA[jj][(4 * fo + m) * 4 + fo];
                float sum = 0.f;
                for (int q = 0; q < 4; ++q)
                    sum += sA[jj][(4 * fo + q) * 4 + fo];
                sum *= 0.25f;                    // IVF_sum[fo, v]
                res = 0.25f * ifv / sum * sW0[t];
            } else {
                res = sIVF[t] * sW0[t];
            }
        }
        out[t] = res;
    }
}

extern "C" void kernel_launch(void* const* d_in, const int* in_sizes, int n_in,
                              void* d_out, int out_size, void* d_ws, size_t ws_size,
                              hipStream_t stream) {
    // setup_inputs order:
    // 0:num_M 1:num_FN 2:num_VN 3:IVF 4:VN_index 5:m 6:n 7:FN_index 8:fa_n 9:w0
    const float* IVF      = (const float*)d_in[3];
    const int*   VN_index = (const int*)  d_in[4];
    const int*   m_s      = (const int*)  d_in[5];
    const int*   n_s      = (const int*)  d_in[6];
    const int*   FN_index = (const int*)  d_in[7];
    const float* fa_n     = (const float*)d_in[8];
    const float* w0       = (const float*)d_in[9];
    float*       out      = (float*)d_out;

    MPA_37056977830474_kernel<<<dim3(1), dim3(32), 0, stream>>>(
        IVF, VN_index, m_s, n_s, FN_index, fa_n, w0, out);
}